// TripletLoss_34849364640098
// MI455X (gfx1250) — compile-verified
//
#include <hip/hip_runtime.h>
#include <hip/hip_bf16.h>
#include <math.h>

typedef __attribute__((ext_vector_type(2))) float v2f;
typedef __attribute__((ext_vector_type(8))) float v8f;

#define TL_N 384
#define TL_D 512
#define TL_MARGIN 0.5f
#define TL_EPS 1e-6f

// ---------------------------------------------------------------------------
// Kernel 1: per-row squared norm q[i] and sum s[i].  grid=384, block=128
// ---------------------------------------------------------------------------
__global__ void tl_rowstats_kernel(const float* __restrict__ F,
                                   float* __restrict__ q,
                                   float* __restrict__ s) {
    __shared__ float redq[128];
    __shared__ float reds[128];
    const int i = blockIdx.x;
    const int t = threadIdx.x;
    float aq = 0.f, as = 0.f;
    const float* row = F + (size_t)i * TL_D;
    for (int d = t; d < TL_D; d += 128) {
        float v = row[d];
        aq = fmaf(v, v, aq);
        as += v;
    }
    redq[t] = aq; reds[t] = as;
    __syncthreads();
    for (int off = 64; off > 0; off >>= 1) {
        if (t < off) { redq[t] += redq[t + off]; reds[t] += reds[t + off]; }
        __syncthreads();
    }
    if (t == 0) { q[i] = redq[0]; s[i] = reds[0]; }
}

// ---------------------------------------------------------------------------
// Kernel 2: Gram matrix via v_wmma_f32_16x16x4_f32, fused into distance.
// 576 waves total (24x24 tiles of 16x16), 4 waves per 128-thread block.
// dist[i][j] = sqrt(q_i + q_j - 2*G_ij + 2*eps*(s_i - s_j) + D*eps^2)
// ---------------------------------------------------------------------------
__global__ void tl_gram_dist_kernel(const float* __restrict__ F,
                                    const float* __restrict__ q,
                                    const float* __restrict__ s,
                                    float* __restrict__ dist) {
    const int wave = (blockIdx.x * blockDim.x + threadIdx.x) >> 5;
    const int lane = threadIdx.x & 31;
    const int tileRow = wave / 24;
    const int tileCol = wave - tileRow * 24;
    const int rowBase = tileRow * 16;
    const int colBase = tileCol * 16;

    const int h = lane >> 4;   // half-wave: selects K pair {0,1} vs {2,3}
    const int m = lane & 15;   // M for A-frag / N for B-frag

    // A fragment: A[m][k] = F[rowBase+m][k]; lane holds K = 2h, 2h+1 per step
    // B fragment: B[k][n] = F[colBase+n][k]; same per-lane addressing pattern
    const float* aPtr = F + (size_t)(rowBase + m) * TL_D + 2 * h;
    const float* bPtr = F + (size_t)(colBase + m) * TL_D + 2 * h;

    v8f c = {};
#pragma unroll 8
    for (int kk = 0; kk < TL_D; kk += 4) {
        v2f a = *(const v2f*)(aPtr + kk);
        v2f b = *(const v2f*)(bPtr + kk);
        // 8 args: (neg_a, A, neg_b, B, c_mod, C, reuse_a, reuse_b)
        c = __builtin_amdgcn_wmma_f32_16x16x4_f32(
                false, a, false, b, (short)0, c, false, false);
    }

    // C/D layout: lane (h,m) VGPR r holds element (M = r + 8h, N = m)
    const int col = colBase + m;
    const float qc = q[col];
    const float sc = s[col];
    const float deps2 = (float)TL_D * TL_EPS * TL_EPS;
#pragma unroll
    for (int r = 0; r < 8; ++r) {
        const int row = rowBase + r + 8 * h;
        float d2 = q[row] + qc - 2.f * c[r] + 2.f * TL_EPS * (s[row] - sc) + deps2;
        dist[(size_t)row * TL_N + col] = sqrtf(fmaxf(d2, 0.f));
    }
}

// ---------------------------------------------------------------------------
// Kernel 3: per-anchor masked hinge sum. grid=384 (one block per anchor),
// block=256. Deterministic: fixed-order strided loops + fixed tree reduce.
// ---------------------------------------------------------------------------
__global__ void tl_hinge_kernel(const float* __restrict__ dist,
                                const int* __restrict__ labels,
                                const int* __restrict__ levels,
                                float* __restrict__ per_anchor) {
    __shared__ float dRow[TL_N];
    __shared__ float posM[TL_N];
    __shared__ float negM[TL_N];
    __shared__ float redf[256];
    __shared__ int   redi[256];

    const int i = blockIdx.x;
    const int t = threadIdx.x;
    const int li = labels[i];
    const int vi = levels[i];

    int cnt_loc = 0;  // packed: (npos<<16) | nneg  (each thread adds <=2 per field)
    for (int j = t; j < TL_N; j += 256) {
        dRow[j] = dist[(size_t)i * TL_N + j];
        const bool sameLab = (labels[j] == li) && (j != i);
        const bool sameLev = (levels[j] == vi);
        const float p = (sameLab && sameLev) ? 1.f : 0.f;
        const float n = (sameLab && !sameLev) ? 1.f : 0.f;
        posM[j] = p;
        negM[j] = n;
        cnt_loc += ((int)p << 16) | (int)n;
    }
    redi[t] = cnt_loc;
    __syncthreads();
    for (int off = 128; off > 0; off >>= 1) {
        if (t < off) redi[t] += redi[t + off];
        __syncthreads();
    }
    const int npos = redi[0] >> 16;
    const int nneg = redi[0] & 0xffff;
    __syncthreads();

    // sum over (j,k) pairs: relu(d_ij - d_ik + margin) * pos[j] * neg[k]
    float acc = 0.f;
    for (int j = 0; j < TL_N; ++j) {
        const float pj = posM[j];
        if (pj == 0.f) continue;  // skipping exact-zero terms: FP-identical
        const float djm = dRow[j] + TL_MARGIN;
        for (int k = t; k < TL_N; k += 256) {
            acc += negM[k] * fmaxf(djm - dRow[k], 0.f);
        }
    }
    redf[t] = acc;
    __syncthreads();
    for (int off = 128; off > 0; off >>= 1) {
        if (t < off) redf[t] += redf[t + off];
        __syncthreads();
    }
    if (t == 0) {
        const float cnt = (float)npos * (float)nneg;
        per_anchor[i] = (cnt > 0.f) ? (redf[0] / cnt) : 0.f;
    }
}

// ---------------------------------------------------------------------------
// Kernel 4: mean over anchors -> d_out[0]. Single block, deterministic tree.
// ---------------------------------------------------------------------------
__global__ void tl_final_kernel(const float* __restrict__ per_anchor,
                                float* __restrict__ out) {
    __shared__ float red[256];
    const int t = threadIdx.x;
    float acc = 0.f;
    for (int i = t; i < TL_N; i += 256) acc += per_anchor[i];
    red[t] = acc;
    __syncthreads();
    for (int off = 128; off > 0; off >>= 1) {
        if (t < off) red[t] += red[t + off];
        __syncthreads();
    }
    if (t == 0) out[0] = red[0] / (float)TL_N;
}

// ---------------------------------------------------------------------------
extern "C" void kernel_launch(void* const* d_in, const int* in_sizes, int n_in,
                              void* d_out, int out_size, void* d_ws, size_t ws_size,
                              hipStream_t stream) {
    (void)in_sizes; (void)n_in; (void)out_size; (void)ws_size;
    const float* F      = (const float*)d_in[0];
    const int*   labels = (const int*)d_in[1];
    const int*   levels = (const int*)d_in[2];
    float*       out    = (float*)d_out;

    float* ws   = (float*)d_ws;
    float* dist = ws;                              // N*N
    float* q    = ws + (size_t)TL_N * TL_N;        // N
    float* s    = q + TL_N;                        // N
    float* pa   = s + TL_N;                        // N

    tl_rowstats_kernel<<<TL_N, 128, 0, stream>>>(F, q, s);

    // 24x24 = 576 output tiles; 4 waves (128 threads) per block -> 144 blocks
    tl_gram_dist_kernel<<<144, 128, 0, stream>>>(F, q, s, dist);

    tl_hinge_kernel<<<TL_N, 256, 0, stream>>>(dist, labels, levels, pa);

    tl_final_kernel<<<1, 256, 0, stream>>>(pa, out);
}